// SGPool_73177652789809
// MI455X (gfx1250) — compile-verified
//
#include <hip/hip_runtime.h>
#include <stdint.h>

#define B_SZ 16
#define N_PTS 32768
#define NPOINT 2048
#define C_FEAT 128
#define THREADS_FPS 1024
#define PPT (N_PTS / THREADS_FPS) /* 32 points per thread, held in VGPRs */

// ---------- CDNA5 async global->LDS capability probes ----------
#if defined(__has_builtin)
#  if __has_builtin(__builtin_amdgcn_global_load_async_to_lds_b128)
#    define USE_ASYNC_LDS 1
#  endif
#  if __has_builtin(__builtin_amdgcn_s_wait_asynccnt)
#    define HAVE_WAIT_ASYNCCNT 1
#  endif
#endif

// Pointer types matching the async builtin's signature:
// void (v4i addrspace(1)*, v4i addrspace(3)*, imm int, imm int)
typedef int v4i_t __attribute__((__vector_size__(16)));
typedef v4i_t __attribute__((address_space(1)))* g_v4i_p;
typedef v4i_t __attribute__((address_space(3)))* l_v4i_p;

// =====================================================================
// Kernel 1: farthest point sampling. One workgroup (32 wave32 waves) per
// batch; all 32768 points live in VGPRs (96 coord regs + 32 dist regs per
// thread). 2048 sequential iterations; block argmax via wave32 shuffle
// reduction + tiny LDS combine (2 barriers / iteration).
// =====================================================================
__global__ __launch_bounds__(THREADS_FPS)
void fps_kernel(const float* __restrict__ xyz, const int* __restrict__ init_far,
                int* __restrict__ out_idx) {
    const int b = blockIdx.x;
    const int tid = threadIdx.x;
    const size_t base = (size_t)b * N_PTS * 3;

    float px[PPT], py[PPT], pz[PPT], dist[PPT];
#pragma unroll
    for (int s = 0; s < PPT; ++s) {
        const size_t p = base + (size_t)(s * THREADS_FPS + tid) * 3;
        px[s] = xyz[p + 0];
        py[s] = xyz[p + 1];
        pz[s] = xyz[p + 2];
        dist[s] = 1e10f;
    }

    __shared__ float s_val[32];
    __shared__ int   s_idx[32];
    __shared__ int   s_win;

    int farthest = init_far[b];
    const int lane = tid & 31;
    const int wv   = tid >> 5;

    for (int it = 0; it < NPOINT; ++it) {
        if (tid == 0) out_idx[b * NPOINT + it] = farthest;  // record pre-update (torch/jax style)

        // centroid fetch: uniform (SGPR) address -> scalar/broadcast load, L2 resident
        const size_t cb = base + (size_t)farthest * 3;
        const float cx = xyz[cb + 0];
        const float cy = xyz[cb + 1];
        const float cz = xyz[cb + 2];

        float bv = -1.0f;
        int bslot = 0;
#pragma unroll
        for (int s = 0; s < PPT; ++s) {
            const float dx = px[s] - cx;
            const float dy = py[s] - cy;
            const float dz = pz[s] - cz;
            const float d  = dx * dx + dy * dy + dz * dz;
            const float nd = fminf(dist[s], d);
            dist[s] = nd;
            if (nd > bv) { bv = nd; bslot = s; }  // strict > keeps lowest index (ascending s)
        }
        int bidx = bslot * THREADS_FPS + tid;     // global point index

        // wave32 (val, min-idx-tie-break) max reduction
#pragma unroll
        for (int off = 16; off > 0; off >>= 1) {
            const float ov = __shfl_xor(bv, off, 32);
            const int   oi = __shfl_xor(bidx, off, 32);
            if (ov > bv || (ov == bv && oi < bidx)) { bv = ov; bidx = oi; }
        }
        if (lane == 0) { s_val[wv] = bv; s_idx[wv] = bidx; }
        __syncthreads();
        if (tid < 32) {  // wave 0 combines the 32 wave partials
            bv   = s_val[tid];
            bidx = s_idx[tid];
#pragma unroll
            for (int off = 16; off > 0; off >>= 1) {
                const float ov = __shfl_xor(bv, off, 32);
                const int   oi = __shfl_xor(bidx, off, 32);
                if (ov > bv || (ov == bv && oi < bidx)) { bv = ov; bidx = oi; }
            }
            if (tid == 0) s_win = bidx;
        }
        __syncthreads();
        farthest = __builtin_amdgcn_readfirstlane(s_win);  // force uniform
    }
}

// =====================================================================
// Kernel 2: gather xyz at sampled indices, write transposed [B,3,NPOINT].
// =====================================================================
__global__ void gather_xyz_kernel(const float* __restrict__ xyz,
                                  const int* __restrict__ idx,
                                  float* __restrict__ out) {
    const int b = blockIdx.y;
    const int s = blockIdx.x * blockDim.x + threadIdx.x;
    if (s >= NPOINT) return;
    const int id = idx[b * NPOINT + s];
    const size_t src = (size_t)b * N_PTS * 3 + (size_t)id * 3;
    const size_t dst = (size_t)b * 3 * NPOINT + s;
#pragma unroll
    for (int k = 0; k < 3; ++k) out[dst + (size_t)k * NPOINT] = xyz[src + k];
}

// =====================================================================
// Kernel 3: gather features + transpose -> [B,128,NPOINT].
// Stages 32 sampled rows (32 x 512B) into LDS with CDNA5 async
// global->LDS b128 copies (ASYNCcnt), then writes coalesced over s.
// =====================================================================
#define TS 32
#define TP 132 /* padded LDS row pitch in floats; 528B keeps 16B alignment */

__global__ __launch_bounds__(256)
void gather_feat_kernel(const float* __restrict__ feat,
                        const int* __restrict__ idx,
                        float* __restrict__ out) {
    __shared__ float tile[TS * TP];
    __shared__ int   sidx[TS];
    const int b   = blockIdx.y;
    const int s0  = blockIdx.x * TS;
    const int tid = threadIdx.x;
    if (tid < TS) sidx[tid] = idx[b * NPOINT + s0 + tid];
    __syncthreads();

    const int lane = tid & 31;  // 32 lanes x 16B = one 512B feature row
    const int grp  = tid >> 5;  // 8 groups, 4 rows each
#if defined(USE_ASYNC_LDS)
#pragma unroll
    for (int j = 0; j < TS / 8; ++j) {
        const int r = grp + 8 * j;
        const float* src = feat + ((size_t)b * N_PTS + (size_t)sidx[r]) * C_FEAT + lane * 4;
        float* dst = &tile[r * TP + lane * 4];
        __builtin_amdgcn_global_load_async_to_lds_b128(
            (g_v4i_p)(uintptr_t)(const void*)src,
            (l_v4i_p)dst,
            0, 0);
    }
#  if defined(HAVE_WAIT_ASYNCCNT)
    __builtin_amdgcn_s_wait_asynccnt(0);
#  else
    asm volatile("s_wait_asynccnt 0" ::: "memory");
#  endif
#else
#pragma unroll
    for (int j = 0; j < TS / 8; ++j) {
        const int r = grp + 8 * j;
        const float4 v =
            *(const float4*)(feat + ((size_t)b * N_PTS + (size_t)sidx[r]) * C_FEAT + lane * 4);
        *(float4*)&tile[r * TP + lane * 4] = v;
    }
#endif
    __syncthreads();

    // transposed write-out: 32 consecutive tids write 32 consecutive s for one c
    const size_t obase = (size_t)b * C_FEAT * NPOINT + s0;
#pragma unroll
    for (int e = 0; e < (C_FEAT * TS) / 256; ++e) {  // 16 elements / thread
        const int linear = tid + e * 256;
        const int c  = linear >> 5;
        const int sl = linear & 31;
        out[obase + (size_t)c * NPOINT + sl] = tile[sl * TP + c];
    }
}

extern "C" void kernel_launch(void* const* d_in, const int* in_sizes, int n_in,
                              void* d_out, int out_size, void* d_ws, size_t ws_size,
                              hipStream_t stream) {
    const float* xyz      = (const float*)d_in[0];  // [16,32768,3]  f32
    const float* feat     = (const float*)d_in[1];  // [16,32768,128] f32
    const int*   init_far = (const int*)d_in[2];    // [16] i32

    float* out_xyz  = (float*)d_out;                                   // [16,3,2048]
    float* out_feat = (float*)d_out + (size_t)B_SZ * 3 * NPOINT;       // [16,128,2048]
    int*   ws_idx   = (int*)d_ws;                                      // [16,2048] i32

    fps_kernel<<<B_SZ, THREADS_FPS, 0, stream>>>(xyz, init_far, ws_idx);

    dim3 gx((NPOINT + 255) / 256, B_SZ);
    gather_xyz_kernel<<<gx, 256, 0, stream>>>(xyz, ws_idx, out_xyz);

    dim3 gf(NPOINT / TS, B_SZ);
    gather_feat_kernel<<<gf, 256, 0, stream>>>(feat, ws_idx, out_feat);
}